// GNN_TransformerConv_14963666059756
// MI455X (gfx1250) — compile-verified
//
#include <hip/hip_runtime.h>
#include <hip/hip_bf16.h>

// ---------------------------------------------------------------------------
// GNN TransformerConv (2 layers) + MLP for MI455X (gfx1250, wave32)
// Dense GEMMs use V_WMMA_F32_16X16X4_F32 (full f32 precision matrix pipe).
// Edge softmax/aggregation uses atomics; edge projection recomputed from LDS.
// ---------------------------------------------------------------------------

typedef float v2f __attribute__((ext_vector_type(2)));
typedef float v8f __attribute__((ext_vector_type(8)));

#define IN_DIM   64
#define HID      32
#define HEADS    2
#define HC       64   // HID*HEADS
#define ED       16
#define OUT_DIM  32
#define NEG_SLOPE 0.01f
#define RSQRT_HID 0.17677669529663687f  // 1/sqrt(32)

// ---------------------------------------------------------------------------
// Generic WMMA f32 GEMM:  Y[M,NOUT] = act( X[M,K] @ W[K,NOUT] + b )
// One wave -> 16 rows x NOUT cols.  K stepped by 4 (16x16x4 f32 WMMA).
// A layout: lane L holds row (L&15); lanes 0-15 carry K=k,k+1; lanes 16-31 K=k+2,k+3.
// B layout (mirror): lane L holds col (L&15) of W rows k+kb, k+kb+1.
// C/D layout: vgpr r -> row r (lanes 0-15) / r+8 (lanes 16-31), col = lane&15.
// ---------------------------------------------------------------------------
template <int K, int NOUT, bool BIAS, bool ACT>
__global__ __launch_bounds__(256) void gemm_wmma_f32(
    const float* __restrict__ X, const float* __restrict__ W,
    const float* __restrict__ B, float* __restrict__ Y, int M)
{
    constexpr int CT = NOUT / 16;
    const int lane = threadIdx.x & 31;
    const int wave = threadIdx.x >> 5;
    const int rowBase = (blockIdx.x * 8 + wave) * 16;
    if (rowBase >= M) return;              // wave-uniform: EXEC stays all-ones

    const int n15 = lane & 15;
    const int kb  = (lane >> 4) * 2;       // 0 or 2
    int rowL = rowBase + n15;              // row this lane loads for A
    if (rowL > M - 1) rowL = M - 1;        // clamp (keeps EXEC full)

    v8f acc[CT];
#pragma unroll
    for (int t = 0; t < CT; ++t) {
        const float bv = BIAS ? B[t * 16 + n15] : 0.0f;
#pragma unroll
        for (int r = 0; r < 8; ++r) acc[t][r] = bv;
    }

    const float* __restrict__ xr = X + (long)rowL * K;
#pragma unroll
    for (int k = 0; k < K; k += 4) {
        v2f a;
        a.x = xr[k + kb];
        a.y = xr[k + kb + 1];
#pragma unroll
        for (int t = 0; t < CT; ++t) {
            v2f b;
            b.x = W[(k + kb)     * NOUT + t * 16 + n15];
            b.y = W[(k + kb + 1) * NOUT + t * 16 + n15];
            acc[t] = __builtin_amdgcn_wmma_f32_16x16x4_f32(
                false, a, false, b, (short)0, acc[t], false, false);
        }
    }

    const int rhalf = (lane >> 4) * 8;
#pragma unroll
    for (int t = 0; t < CT; ++t) {
#pragma unroll
        for (int r = 0; r < 8; ++r) {
            const int row = rowBase + rhalf + r;
            if (row < M) {
                float v = acc[t][r];
                if (ACT) v = (v >= 0.0f) ? v : NEG_SLOPE * v;
                Y[(long)row * NOUT + t * 16 + n15] = v;
            }
        }
    }
}

// ---------------------------------------------------------------------------
// Order-preserving float <-> uint mapping for atomic segment-max
// ---------------------------------------------------------------------------
__device__ __forceinline__ unsigned f2ord(float f) {
    unsigned u = __float_as_uint(f);
    return (u & 0x80000000u) ? ~u : (u | 0x80000000u);
}
__device__ __forceinline__ float ord2f(unsigned u) {
    return (u & 0x80000000u) ? __uint_as_float(u ^ 0x80000000u)
                             : __uint_as_float(~u);
}
#define ORD_NEG_INF 0x007FFFFFu   // f2ord(-inf)

__global__ __launch_bounds__(256) void fill_u32(unsigned* p, unsigned v, int n) {
    int i = blockIdx.x * blockDim.x + threadIdx.x;
    if (i < n) p[i] = v;
}

// ---------------------------------------------------------------------------
// Pass A: per edge, both heads: alpha = q[dst] . (k[src] + ea@We) / sqrt(32)
//         store alpha, atomic segment-max into amaxU[dst,h]
// ---------------------------------------------------------------------------
__global__ __launch_bounds__(256) void edge_alpha(
    const int* __restrict__ src, const int* __restrict__ dst,
    const float* __restrict__ ea, const float* __restrict__ q,
    const float* __restrict__ k,  const float* __restrict__ We,
    float* __restrict__ alpha, unsigned* __restrict__ amaxU, int E)
{
    __shared__ float sWe[ED * HC];
    for (int i = threadIdx.x; i < ED * HC; i += blockDim.x) sWe[i] = We[i];
    __syncthreads();

    const int e = blockIdx.x * blockDim.x + threadIdx.x;
    if (e >= E) return;
    const int s = src[e], d = dst[e];

    float av[ED];
#pragma unroll
    for (int t = 0; t < ED; ++t) av[t] = ea[(long)e * ED + t];

    const float* __restrict__ qd = q + (long)d * HC;
    const float* __restrict__ ks = k + (long)s * HC;

#pragma unroll
    for (int h = 0; h < HEADS; ++h) {
        float dot = 0.0f;
#pragma unroll 4
        for (int c = 0; c < HID; ++c) {
            float ep = 0.0f;
#pragma unroll
            for (int t = 0; t < ED; ++t)
                ep = fmaf(av[t], sWe[t * HC + h * HID + c], ep);
            dot = fmaf(qd[h * HID + c], ks[h * HID + c] + ep, dot);
        }
        const float al = dot * RSQRT_HID;
        alpha[(long)e * HEADS + h] = al;
        atomicMax(&amaxU[d * HEADS + h], f2ord(al));
    }
}

// ---------------------------------------------------------------------------
// Pass B: ex = exp(alpha - amax[dst]); alpha <- ex; denom[dst,h] += ex
// ---------------------------------------------------------------------------
__global__ __launch_bounds__(256) void edge_expsum(
    const int* __restrict__ dst, float* __restrict__ alpha,
    const unsigned* __restrict__ amaxU, float* __restrict__ denom, int E2)
{
    const int i = blockIdx.x * blockDim.x + threadIdx.x;
    if (i >= E2) return;
    const int e = i >> 1, h = i & 1;
    const int d = dst[e];
    const float m  = ord2f(amaxU[d * HEADS + h]);
    const float ex = __expf(alpha[i] - m);
    alpha[i] = ex;
    atomicAdd(&denom[d * HEADS + h], ex);
}

// ---------------------------------------------------------------------------
// Pass C: coef = ex / (denom[dst]+eps); out[dst] += (v[src] + ea@We) * coef
// ---------------------------------------------------------------------------
__global__ __launch_bounds__(256) void edge_scatter(
    const int* __restrict__ src, const int* __restrict__ dst,
    const float* __restrict__ ea, const float* __restrict__ v,
    const float* __restrict__ We, const float* __restrict__ exv,
    const float* __restrict__ denom, float* __restrict__ out, int E2)
{
    __shared__ float sWe[ED * HC];
    for (int i = threadIdx.x; i < ED * HC; i += blockDim.x) sWe[i] = We[i];
    __syncthreads();

    const int i = blockIdx.x * blockDim.x + threadIdx.x;
    if (i >= E2) return;
    const int e = i >> 1, h = i & 1;
    const int s = src[e], d = dst[e];
    const float coef = exv[i] / (denom[d * HEADS + h] + 1e-16f);

    float av[ED];
#pragma unroll
    for (int t = 0; t < ED; ++t) av[t] = ea[(long)e * ED + t];

    const float* __restrict__ vs = v + (long)s * HC + h * HID;
    float* __restrict__ od = out + (long)d * HC + h * HID;
#pragma unroll 4
    for (int c = 0; c < HID; ++c) {
        float ep = 0.0f;
#pragma unroll
        for (int t = 0; t < ED; ++t)
            ep = fmaf(av[t], sWe[t * HC + h * HID + c], ep);
        atomicAdd(&od[c], (vs[c] + ep) * coef);
    }
}

__global__ __launch_bounds__(256) void leaky_k(const float* __restrict__ in,
                                               float* __restrict__ out, int n) {
    int i = blockIdx.x * blockDim.x + threadIdx.x;
    if (i < n) {
        float z = in[i];
        out[i] = (z >= 0.0f) ? z : NEG_SLOPE * z;
    }
}

// ---------------------------------------------------------------------------
// Host orchestration
// ---------------------------------------------------------------------------
static inline int cdiv(int a, int b) { return (a + b - 1) / b; }

extern "C" void kernel_launch(void* const* d_in, const int* in_sizes, int n_in,
                              void* d_out, int out_size, void* d_ws, size_t ws_size,
                              hipStream_t stream) {
    const int N = in_sizes[0] / IN_DIM;
    const int E = in_sizes[2] / 2;

    const float* x    = (const float*)d_in[0];
    const int*   eidx = (const int*)  d_in[2];
    const float* ea   = (const float*)d_in[3];
    const int* srcp = eidx;
    const int* dstp = eidx + E;

    // per-layer params: Wq,bq,Wk,bk,Wv,bv,We,Ws,bs starting at index 5 and 14
    const float* P[2][9];
    for (int l = 0; l < 2; ++l)
        for (int j = 0; j < 9; ++j)
            P[l][j] = (const float*)d_in[5 + l * 9 + j];
    const float* Wm1 = (const float*)d_in[23];
    const float* bm1 = (const float*)d_in[24];
    const float* Wm2 = (const float*)d_in[25];
    const float* bm2 = (const float*)d_in[26];

    // workspace carve-up (floats)
    float* ws = (float*)d_ws;
    float*    q     = ws;                 // N*HC
    float*    kbuf  = q     + (long)N * HC;
    float*    vbuf  = kbuf  + (long)N * HC;
    float*    accb  = vbuf  + (long)N * HC;   // skip + aggregated messages
    float*    xbuf  = accb  + (long)N * HC;   // activated layer output
    float*    alpha = xbuf  + (long)N * HC;   // E*HEADS (alpha then ex)
    unsigned* amaxU = (unsigned*)(alpha + (long)E * HEADS);
    float*    denom = (float*)(amaxU + (long)N * HEADS);
    float*    h1    = denom + (long)N * HEADS; // N*HID

    const int gemmGrid = cdiv(N, 128);        // 8 waves * 16 rows per block
    const int eGrid    = cdiv(E, 256);
    const int e2Grid   = cdiv(E * HEADS, 256);
    const int nhGrid   = cdiv(N * HEADS, 256);
    const int nfGrid   = cdiv(N * HC, 256);

    const float* xin = x;
    for (int l = 0; l < 2; ++l) {
        const float *Wq = P[l][0], *bq = P[l][1], *Wk = P[l][2], *bk = P[l][3];
        const float *Wv = P[l][4], *bv = P[l][5], *We = P[l][6];
        const float *Ws = P[l][7], *bs = P[l][8];

        // node projections (f32 WMMA)
        gemm_wmma_f32<IN_DIM, HC, true, false><<<gemmGrid, 256, 0, stream>>>(xin, Wq, bq, q,    N);
        gemm_wmma_f32<IN_DIM, HC, true, false><<<gemmGrid, 256, 0, stream>>>(xin, Wk, bk, kbuf, N);
        gemm_wmma_f32<IN_DIM, HC, true, false><<<gemmGrid, 256, 0, stream>>>(xin, Wv, bv, vbuf, N);
        gemm_wmma_f32<IN_DIM, HC, true, false><<<gemmGrid, 256, 0, stream>>>(xin, Ws, bs, accb, N);

        // init segment-max / segment-sum buffers
        fill_u32<<<nhGrid, 256, 0, stream>>>(amaxU, ORD_NEG_INF, N * HEADS);
        fill_u32<<<nhGrid, 256, 0, stream>>>((unsigned*)denom, 0u, N * HEADS);

        // edge attention passes
        edge_alpha  <<<eGrid,  256, 0, stream>>>(srcp, dstp, ea, q, kbuf, We, alpha, amaxU, E);
        edge_expsum <<<e2Grid, 256, 0, stream>>>(dstp, alpha, amaxU, denom, E * HEADS);
        edge_scatter<<<e2Grid, 256, 0, stream>>>(srcp, dstp, ea, vbuf, We, alpha, denom, accb, E * HEADS);

        // activation -> next layer input
        leaky_k<<<nfGrid, 256, 0, stream>>>(accb, xbuf, N * HC);
        xin = xbuf;
    }

    // MLP head: h1 = leaky(x@Wm1+bm1) ; out = h1@Wm2+bm2
    gemm_wmma_f32<HC,  HID,     true, true ><<<gemmGrid, 256, 0, stream>>>(xbuf, Wm1, bm1, h1, N);
    gemm_wmma_f32<HID, OUT_DIM, true, false><<<gemmGrid, 256, 0, stream>>>(h1,  Wm2, bm2, (float*)d_out, N);
}